// GraphAttentionRela_73701638799908
// MI455X (gfx1250) — compile-verified
//
#include <hip/hip_runtime.h>

// ---------------- problem constants ----------------
#define B_     16
#define ATT_   64
#define RELA_  128
#define RNN_   1024
#define HID_   512
#define BATT_  (B_ * ATT_)   // 1024 rows of the first GEMM
#define EPS_   1e-8f

typedef __attribute__((ext_vector_type(2))) float v2f;
typedef __attribute__((ext_vector_type(8))) float v8f;

// ---------------------------------------------------------------------
// Native TRANS-unit tanh (V_TANH_F32 on gfx1250). Prefer the clang
// builtin (compiler tracks the TRANS data hazard); otherwise emit the
// instruction directly with a trailing v_nop per ISA §7.4 (one
// independent op / V_NOP required before the TRANS result is consumed).
// ---------------------------------------------------------------------
__device__ __forceinline__ float htanh(float x) {
#if __has_builtin(__builtin_amdgcn_tanhf)
    return __builtin_amdgcn_tanhf(x);
#elif __has_builtin(__builtin_amdgcn_tanh_f32)
    return __builtin_amdgcn_tanh_f32(x);
#else
    float y;
    asm volatile("v_tanh_f32 %0, %1\n\tv_nop" : "=v"(y) : "v"(x));
    return y;
#endif
}

// =====================================================================
// Kernel 1: att_obj[1024,512] = obj_vecs_e[1024,1024] @ W_obj[1024,512] + b_obj
// fp32 WMMA 16x16x4. One wave per 16(M) x 64(N) strip; A fragment reused
// across the 4 N-tiles of the strip. 512 waves total = 64 blocks x 8 waves.
// =====================================================================
__global__ __launch_bounds__(256) void gemm1_kernel(
    const float* __restrict__ A,     // [1024, 1024]
    const float* __restrict__ W,     // [1024, 512]
    const float* __restrict__ bias,  // [512]
    float* __restrict__ C)           // [1024, 512]
{
    const int wave = (blockIdx.x * blockDim.x + threadIdx.x) >> 5;  // 0..511
    const int lane = threadIdx.x & 31;
    const int mt   = wave >> 3;       // 64 M-tiles
    const int ns   = wave & 7;        // 8 N-strips of 64
    const int row0 = mt * 16;
    const int col0 = ns * 64;
    const int half = lane >> 4;       // 0: K pair {0,1}, 1: K pair {2,3}
    const int mr   = lane & 15;

    v8f acc[4] = {};
    for (int k = 0; k < RNN_; k += 4) {
        // A fragment (16x4): lane<16 -> K=k,k+1 ; lane>=16 -> K=k+2,k+3
        const float* ap = A + (size_t)(row0 + mr) * RNN_ + k + half * 2;
        v2f a;  a.x = ap[0];  a.y = ap[1];
#pragma unroll
        for (int t = 0; t < 4; ++t) {
            const int n0 = col0 + t * 16;
            // B fragment (4x16): lane<16 -> rows k,k+1 ; lane>=16 -> rows k+2,k+3
            const float* bp = W + (size_t)(k + half * 2) * HID_ + n0 + mr;
            v2f b;  b.x = bp[0];  b.y = bp[HID_];
            acc[t] = __builtin_amdgcn_wmma_f32_16x16x4_f32(
                false, a, false, b, (short)0, acc[t], false, false);
        }
    }
#pragma unroll
    for (int t = 0; t < 4; ++t) {
        const int n0 = col0 + t * 16;
        const float bv = bias[n0 + mr];
#pragma unroll
        for (int v = 0; v < 8; ++v) {
            const int row = row0 + v + half * 8;   // C/D layout: M=v (lanes 0-15), M=v+8 (16-31)
            C[(size_t)row * HID_ + n0 + mr] = acc[t][v] + bv;
        }
    }
}

// =====================================================================
// Kernel 2: fused scores + softmax + mask + renorm  -> weight [16,64,128]
// One block (8 waves) per (b,a). Avoids materializing the 256 MB `dot`
// tensor entirely: att_obj row + w_alpha live in LDS, tanh/FMA streamed
// through the native V_TANH_F32 TRANS unit.
// =====================================================================
__global__ __launch_bounds__(256) void attn_scores_kernel(
    const float* __restrict__ att_obj,   // [1024, 512]
    const float* __restrict__ p_rela,    // [16, 128, 512]
    const float* __restrict__ w_alpha,   // [512]
    const float* __restrict__ b_alpha_p, // [1]
    const int*   __restrict__ mask,      // [16, 64, 128]
    float* __restrict__ weight)          // [16, 64, 128]
{
    __shared__ float s_att[HID_];
    __shared__ float s_wa[HID_];
    __shared__ float s_sc[RELA_];
    __shared__ float s_red[2];

    const int ba   = blockIdx.x;       // b*64 + a
    const int b    = ba >> 6;
    const int tid  = threadIdx.x;
    const int lane = tid & 31;
    const int wv   = tid >> 5;

    for (int h = tid; h < HID_; h += 256) {
        s_att[h] = att_obj[(size_t)ba * HID_ + h];
        s_wa[h]  = w_alpha[h];
    }
    __syncthreads();

    // scores: each wave handles 16 relations; lanes stride the HID axis.
    const float* prow_base = p_rela + (size_t)b * RELA_ * HID_;
#pragma unroll 1
    for (int i = 0; i < 16; ++i) {
        const int r = wv * 16 + i;
        const float* pr = prow_base + (size_t)r * HID_;
        float acc = 0.f;
#pragma unroll 4
        for (int h = lane; h < HID_; h += 32)
            acc = fmaf(s_wa[h], htanh(pr[h] + s_att[h]), acc);
        acc += __shfl_down(acc, 16);
        acc += __shfl_down(acc, 8);
        acc += __shfl_down(acc, 4);
        acc += __shfl_down(acc, 2);
        acc += __shfl_down(acc, 1);
        if (lane == 0) s_sc[r] = acc + b_alpha_p[0];
    }
    __syncthreads();

    // softmax max (wave 0 reduces all 128 scores)
    if (wv == 0) {
        float m = fmaxf(fmaxf(s_sc[lane], s_sc[lane + 32]),
                        fmaxf(s_sc[lane + 64], s_sc[lane + 96]));
        m = fmaxf(m, __shfl_down(m, 16));
        m = fmaxf(m, __shfl_down(m, 8));
        m = fmaxf(m, __shfl_down(m, 4));
        m = fmaxf(m, __shfl_down(m, 2));
        m = fmaxf(m, __shfl_down(m, 1));
        if (lane == 0) s_red[0] = m;
    }
    __syncthreads();
    const float mv = s_red[0];
    if (tid < RELA_) s_sc[tid] = __expf(s_sc[tid] - mv);
    __syncthreads();

    // exp-sum
    if (wv == 0) {
        float s = s_sc[lane] + s_sc[lane + 32] + s_sc[lane + 64] + s_sc[lane + 96];
        s += __shfl_down(s, 16);
        s += __shfl_down(s, 8);
        s += __shfl_down(s, 4);
        s += __shfl_down(s, 2);
        s += __shfl_down(s, 1);
        if (lane == 0) s_red[1] = s;
    }
    __syncthreads();
    const float esum = s_red[1];

    // mask + renormalize exactly as reference: p = e/esum; pm = p*mask; pm/(sum(pm)+EPS)
    float wm = 0.f;
    if (tid < RELA_) {
        wm = (s_sc[tid] / esum) * (float)mask[(size_t)ba * RELA_ + tid];
        s_sc[tid] = wm;
    }
    __syncthreads();
    if (wv == 0) {
        float s = s_sc[lane] + s_sc[lane + 32] + s_sc[lane + 64] + s_sc[lane + 96];
        s += __shfl_down(s, 16);
        s += __shfl_down(s, 8);
        s += __shfl_down(s, 4);
        s += __shfl_down(s, 2);
        s += __shfl_down(s, 1);
        if (lane == 0) s_red[0] = s;
    }
    __syncthreads();
    if (tid < RELA_)
        weight[(size_t)ba * RELA_ + tid] = wm / (s_red[0] + EPS_);
}

// =====================================================================
// Kernel 3: att_res[b] (64x1024) = weight[b] (64x128) @ rela_feats[b] (128x1024)
// fp32 WMMA 16x16x4, wave per 16(M) x 64(N) strip. 1024 waves = 128 blocks.
// =====================================================================
__global__ __launch_bounds__(256) void gemm2_kernel(
    const float* __restrict__ Wt,   // weight [16, 64, 128]
    const float* __restrict__ R,    // rela_feats [16, 128, 1024]
    float* __restrict__ out)        // [16, 64, 1024]
{
    const int wave = (blockIdx.x * blockDim.x + threadIdx.x) >> 5;  // 0..1023
    const int lane = threadIdx.x & 31;
    const int b    = wave >> 6;
    const int rem  = wave & 63;
    const int mt   = rem >> 4;      // 4 M-tiles (a)
    const int ns   = rem & 15;      // 16 N-strips of 64 (d)
    const int row0 = mt * 16;
    const int col0 = ns * 64;
    const int half = lane >> 4;
    const int mr   = lane & 15;

    const float* A  = Wt + (size_t)b * ATT_ * RELA_;
    const float* Bm = R  + (size_t)b * RELA_ * RNN_;
    float* C = out + (size_t)b * ATT_ * RNN_;

    v8f acc[4] = {};
    for (int k = 0; k < RELA_; k += 4) {
        const float* ap = A + (size_t)(row0 + mr) * RELA_ + k + half * 2;
        v2f a;  a.x = ap[0];  a.y = ap[1];
#pragma unroll
        for (int t = 0; t < 4; ++t) {
            const float* bp = Bm + (size_t)(k + half * 2) * RNN_ + col0 + t * 16 + mr;
            v2f bb;  bb.x = bp[0];  bb.y = bp[RNN_];
            acc[t] = __builtin_amdgcn_wmma_f32_16x16x4_f32(
                false, a, false, bb, (short)0, acc[t], false, false);
        }
    }
#pragma unroll
    for (int t = 0; t < 4; ++t) {
        const int n0 = col0 + t * 16;
#pragma unroll
        for (int v = 0; v < 8; ++v) {
            const int row = row0 + v + half * 8;
            C[(size_t)row * RNN_ + n0 + mr] = acc[t][v];
        }
    }
}

// =====================================================================
// Host launcher
// =====================================================================
extern "C" void kernel_launch(void* const* d_in, const int* in_sizes, int n_in,
                              void* d_out, int out_size, void* d_ws, size_t ws_size,
                              hipStream_t stream) {
    (void)in_sizes; (void)n_in; (void)out_size; (void)ws_size;

    const float* obj_vecs_e = (const float*)d_in[0];  // [16,64,1024]
    const float* rela_feats = (const float*)d_in[1];  // [16,128,1024]
    const float* p_rela     = (const float*)d_in[2];  // [16,128,512]
    const float* W_obj      = (const float*)d_in[3];  // [1024,512]
    const float* b_obj      = (const float*)d_in[4];  // [512]
    const float* w_alpha    = (const float*)d_in[5];  // [512]
    const float* b_alpha    = (const float*)d_in[6];  // [1]
    const int*   masks      = (const int*)d_in[7];    // [16,64,128]
    float* att_res = (float*)d_out;                   // [16,64,1024]

    // workspace layout
    float* att_obj = (float*)d_ws;                    // 1024*512 floats (2 MB)
    float* weight  = att_obj + (size_t)BATT_ * HID_;  // 16*64*128 floats (512 KB)

    // 1) att_obj = obj_vecs_e @ W_obj + b_obj   (512 waves)
    gemm1_kernel<<<64, 256, 0, stream>>>(obj_vecs_e, W_obj, b_obj, att_obj);

    // 2) fused tanh-scores + softmax + mask + renorm (block per (b,a))
    attn_scores_kernel<<<BATT_, 256, 0, stream>>>(att_obj, p_rela, w_alpha,
                                                  b_alpha, masks, weight);

    // 3) att_res = weight @ rela_feats (batched, 1024 waves)
    gemm2_kernel<<<128, 256, 0, stream>>>(weight, rela_feats, att_res);
}